// close_33775622816249
// MI455X (gfx1250) — compile-verified
//
#include <hip/hip_runtime.h>
#include <math.h>

typedef __attribute__((ext_vector_type(2))) float v2f;
typedef __attribute__((ext_vector_type(8))) float v8f;

#define BATCH 8
#define NPTS  4096
#define LFEAT 24
#define KTOP  16
#define NTILE 256          // NPTS / 16 column tiles
#define WAVES_PER_BLOCK 8

// Wave-local LDS ordering point: LDS ops from one wave are processed in order
// (CDNA5 ISA ch.11), so for a per-wave LDS region a DScnt drain + an IR-level
// memory ordering point replaces a block-wide s_barrier.
#define WAVE_LDS_FENCE() __asm__ volatile("s_wait_dscnt 0" ::: "memory")

// ---------------- Phase 1: inverse norms ----------------
// inv[b*NPTS + n] = 1 / (||feat[b,n,:]|| + 1e-8),  feat[b,n,l] = x_c[b,l,0,n]
__global__ __launch_bounds__(256) void invnorm_kernel(const float* __restrict__ xc,
                                                      float* __restrict__ inv) {
    int t = blockIdx.x * blockDim.x + threadIdx.x;   // 0 .. BATCH*NPTS-1
    int b = t >> 12;
    int n = t & (NPTS - 1);
    const float* xb = xc + (size_t)b * LFEAT * 2 * NPTS + n;
    float ss = 0.0f;
#pragma unroll
    for (int l = 0; l < LFEAT; ++l) {
        float f = xb[(size_t)l * 2 * NPTS];
        ss += f * f;
    }
    inv[t] = 1.0f / (sqrtf(ss) + 1e-8f);
}

// ---------------- Phase 2: WMMA Gram + streaming top-16 + gather ----------------
__global__ __launch_bounds__(256) void knn_wmma_kernel(const float* __restrict__ xc,
                                                       const float* __restrict__ inv,
                                                       float* __restrict__ out) {
    __shared__ float tile[WAVES_PER_BLOCK][16][16];   // 8 KB: per-wave D tile
    __shared__ float mval[WAVES_PER_BLOCK][16][KTOP]; // 8 KB: per-wave merge values
    __shared__ int   midx[WAVES_PER_BLOCK][16][KTOP]; // 8 KB: per-wave merge indices

    const int w    = threadIdx.x >> 5;   // wave in block
    const int lane = threadIdx.x & 31;
    const int half = lane >> 4;          // 0: lanes 0-15, 1: lanes 16-31
    const int lr   = lane & 15;

    const int tileId = blockIdx.x * WAVES_PER_BLOCK + w;  // 0 .. 2047
    const int b  = tileId >> 8;                            // 256 row-tiles per batch
    const int n0 = (tileId & (NTILE - 1)) << 4;

    const float* xb   = xc  + (size_t)b * LFEAT * 2 * NPTS;  // [l][0][n] at xb[l*2*NPTS + n]
    const float* invb = inv + b * NPTS;

    // A fragment: rows n0..n0+15, K=24 in 6 chunks of 4.
    // ISA f32 16x4 A layout: lanes 0-15 hold (K=0,K=1) in (v0,v1); lanes 16-31 hold (K=2,K=3).
    v2f afrag[6];
    {
        const int row = n0 + lr;
        const float ir = invb[row];
#pragma unroll
        for (int c = 0; c < 6; ++c) {
            const int k = 4 * c + 2 * half;
            afrag[c].x = xb[(size_t)k       * 2 * NPTS + row] * ir;
            afrag[c].y = xb[(size_t)(k + 1) * 2 * NPTS + row] * ir;
        }
    }

    // streaming top-16 state (dynamic indexing only on the rare replace path)
    float tv[KTOP];
    int   ti[KTOP];
#pragma unroll
    for (int j = 0; j < KTOP; ++j) { tv[j] = -3.402823466e38f; ti[j] = 0x7fffffff; }
    float tmin = -3.402823466e38f;
    int   tminpos = 0;

    for (int mt = 0; mt < NTILE; ++mt) {
        const int m0 = mt << 4;

        // B fragment: symmetric Gram -> identical gather pattern on columns m0..m0+15
        v2f bfrag[6];
        {
            const int col = m0 + lr;
            const float ic = invb[col];
#pragma unroll
            for (int c = 0; c < 6; ++c) {
                const int k = 4 * c + 2 * half;
                bfrag[c].x = xb[(size_t)k       * 2 * NPTS + col] * ic;
                bfrag[c].y = xb[(size_t)(k + 1) * 2 * NPTS + col] * ic;
            }
        }

        // 6 accumulating f32 WMMAs cover K=24 exactly
        v8f acc = {};
#pragma unroll
        for (int c = 0; c < 6; ++c) {
            acc = __builtin_amdgcn_wmma_f32_16x16x4_f32(
                false, afrag[c], false, bfrag[c], (short)0, acc, false, false);
        }

        // order previous iteration's scan reads before overwriting the tile
        WAVE_LDS_FENCE();
        // D layout: VGPR v -> row v (lanes 0-15) / row v+8 (lanes 16-31), col = lane&15
#pragma unroll
        for (int v = 0; v < 8; ++v) {
            tile[w][v + 8 * half][lr] = acc[v];
        }
        // wave-local store -> cross-lane read handoff
        WAVE_LDS_FENCE();

        // scan: lane owns row lr, 8 consecutive columns starting at 8*half.
        // Bulk-load first so the compiler can merge into wide ds_loads.
        const int cbase = 8 * half;
        float vals[8];
#pragma unroll
        for (int c = 0; c < 8; ++c) {
            vals[c] = tile[w][lr][cbase + c];
        }
#pragma unroll
        for (int c = 0; c < 8; ++c) {
            const float val = vals[c];
            if (val > tmin) {
                tv[tminpos] = val;
                ti[tminpos] = m0 + cbase + c;
                tmin = tv[0]; tminpos = 0;
#pragma unroll
                for (int j = 1; j < KTOP; ++j) {
                    if (tv[j] < tmin) { tmin = tv[j]; tminpos = j; }
                }
            }
        }
    }

    // merge upper-half candidates into lower-half lane (wave-local, per-wave LDS)
    if (half == 1) {
#pragma unroll
        for (int j = 0; j < KTOP; ++j) {
            mval[w][lr][j] = tv[j];
            midx[w][lr][j] = ti[j];
        }
    }
    WAVE_LDS_FENCE();

    if (half == 0) {
#pragma unroll
        for (int j = 0; j < KTOP; ++j) {
            const float val = mval[w][lr][j];
            const int   gi  = midx[w][lr][j];
            if (val > tmin) {
                tv[tminpos] = val;
                ti[tminpos] = gi;
                tmin = tv[0]; tminpos = 0;
#pragma unroll
                for (int k = 1; k < KTOP; ++k) {
                    if (tv[k] < tmin) { tmin = tv[k]; tminpos = k; }
                }
            }
        }

        // sort descending by value, ties by lower index (jax.lax.top_k order)
        for (int i = 1; i < KTOP; ++i) {
            const float v = tv[i];
            const int   x = ti[i];
            int j = i - 1;
            while (j >= 0 && (tv[j] < v || (tv[j] == v && ti[j] > x))) {
                tv[j + 1] = tv[j]; ti[j + 1] = ti[j]; --j;
            }
            tv[j + 1] = v; ti[j + 1] = x;
        }

        // output: [feat | 16 gathered neighbors | mean of the 17]
        const int n = n0 + lr;
        float* obase = out + ((size_t)(b * NPTS + n)) * LFEAT * 18;
        for (int l = 0; l < LFEAT; ++l) {
            const float* xl = xb + (size_t)l * 2 * NPTS;
            const float f = xl[n];
            float s = f;
            float* o = obase + (size_t)l * 18;
            o[0] = f;
#pragma unroll
            for (int j = 0; j < KTOP; ++j) {
                const float sv = xl[ti[j]];
                o[1 + j] = sv;
                s += sv;
            }
            o[17] = s * (1.0f / 17.0f);
        }
    }
}

extern "C" void kernel_launch(void* const* d_in, const int* in_sizes, int n_in,
                              void* d_out, int out_size, void* d_ws, size_t ws_size,
                              hipStream_t stream) {
    (void)in_sizes; (void)n_in; (void)out_size; (void)ws_size;
    const float* xc  = (const float*)d_in[0];
    float*       inv = (float*)d_ws;          // BATCH*NPTS floats = 128 KB
    float*       out = (float*)d_out;

    hipLaunchKernelGGL(invnorm_kernel, dim3((BATCH * NPTS) / 256), dim3(256), 0, stream,
                       xc, inv);
    hipLaunchKernelGGL(knn_wmma_kernel, dim3((BATCH * NTILE) / WAVES_PER_BLOCK), dim3(256), 0, stream,
                       xc, inv, out);
}